// DyTraid_88545045774491
// MI455X (gfx1250) — compile-verified
//
#include <hip/hip_runtime.h>
#include <math.h>

typedef __attribute__((ext_vector_type(2))) float v2f;
typedef __attribute__((ext_vector_type(8))) float v8f;
typedef __attribute__((ext_vector_type(4))) int   v4i;

#define D 64
#define DP (D + 4)   // row stride in floats: 272B, 16B-aligned rows, conflict-free reads
#define SPLIT 4
#define CPI 128      // columns per iteration per workgroup (8 waves x 16 cols)

#ifndef __has_builtin
#define __has_builtin(x) 0
#endif

#if __has_builtin(__builtin_amdgcn_global_load_async_to_lds_b128) && \
    __has_builtin(__builtin_amdgcn_s_wait_asynccnt)
#define USE_ASYNC 1
#else
#define USE_ASYNC 0
#endif

#if USE_ASYNC
typedef __attribute__((address_space(1))) v4i gv4i;   // global int4
typedef __attribute__((address_space(3))) v4i lv4i;   // LDS int4
__device__ __forceinline__ void async_copy16(const void* g, void* l) {
    // 16B global -> LDS, tracked by ASYNCcnt, no VGPR data movement
    __builtin_amdgcn_global_load_async_to_lds_b128((gv4i*)g, (lv4i*)l, 0, 0);
}
#endif

__device__ __forceinline__ float softplusf(float x) {
    return (x > 20.f) ? x : log1pf(expf(x));
}

// ---------------- init: zero accumulators / diag / logsum ----------------
__global__ void k_init(float* ws, int n) {
    int i = blockIdx.x * blockDim.x + threadIdx.x;
    if (i < n) ws[i] = 0.f;
}

// ---------------- diag[l] = theta_l . u_l ; smooth = sum (u - uprev)^2 ----
__global__ void k_prep(const float* __restrict__ emb, const float* __restrict__ prev,
                       const float* __restrict__ theta, float* __restrict__ diag,
                       float* acc_smooth, int N) {
    int wave = threadIdx.x >> 5;
    int lane = threadIdx.x & 31;
    int row = blockIdx.x * 8 + wave;
    if (row >= N) return;
    const float* e = emb   + (size_t)row * D;
    const float* p = prev  + (size_t)row * D;
    const float* t = theta + (size_t)row * D;
    float d = 0.f, sm = 0.f;
    for (int c = lane; c < D; c += 32) {
        float ev = e[c];
        float dv = ev - p[c];
        d  += t[c] * ev;
        sm += dv * dv;
    }
    for (int off = 16; off >= 1; off >>= 1) {
        d  += __shfl_xor(d,  off, 32);
        sm += __shfl_xor(sm, off, 32);
    }
    if (lane == 0) { diag[row] = d; atomicAdd(acc_smooth, sm); }
}

// ---------------- L_sh triplet hinge ----------------
__global__ void k_triplet(const float* __restrict__ emb, const int* __restrict__ jI,
                          const int* __restrict__ kI, const int* __restrict__ nI,
                          float* acc_sh, int S) {
    int wave = threadIdx.x >> 5;
    int lane = threadIdx.x & 31;
    int s = blockIdx.x * 8 + wave;
    if (s >= S) return;
    const float* uj = emb + (size_t)jI[s] * D;
    const float* uk = emb + (size_t)kI[s] * D;
    const float* un = emb + (size_t)nI[s] * D;
    float pos = 0.f, neg = 0.f;
    for (int c = lane; c < D; c += 32) {
        float kv = uk[c];
        float a = uj[c] - kv; pos += a * a;
        float b = un[c] - kv; neg += b * b;
    }
    for (int off = 16; off >= 1; off >>= 1) {
        pos += __shfl_xor(pos, off, 32);
        neg += __shfl_xor(neg, off, 32);
    }
    if (lane == 0) {
        float h = pos - neg + 1.0f;   // MARGIN = 1.0
        if (h > 0.f) atomicAdd(acc_sh, h);
    }
}

// ---------------- main: dual GEMM (WMMA f32) + masked log-product ----------
__global__ __launch_bounds__(256) void k_main(
    const float* __restrict__ emb, const float* __restrict__ theta,
    const float* __restrict__ adj, const float* __restrict__ diag,
    const int* __restrict__ iI, const int* __restrict__ jI, const int* __restrict__ kI,
    float* __restrict__ logsum, float* __restrict__ dotk, int N) {

    __shared__ float sUi[16][DP];
    __shared__ float sUj[16][DP];
    __shared__ float sTh[CPI][DP];
    __shared__ int   sRi[16], sRj[16], sK[16];

    const int tid   = threadIdx.x;
    const int sBase = (blockIdx.x / SPLIT) * 16;
    const int split =  blockIdx.x % SPLIT;
    const int colsPerSplit = N / SPLIT;
    const int colStart = split * colsPerSplit;

    if (tid < 16) {
        sRi[tid] = iI[sBase + tid];
        sRj[tid] = jI[sBase + tid];
        sK[tid]  = kI[sBase + tid];
    }
    __syncthreads();

    // ---- stage Ui / Uj (16 x 64) into LDS ----
#if USE_ASYNC
    {
        int r = tid >> 4;            // 256 threads -> one 16B chunk per thread
        int c = (tid & 15) * 4;
        async_copy16(emb + (size_t)sRi[r] * D + c, &sUi[r][c]);
        async_copy16(emb + (size_t)sRj[r] * D + c, &sUj[r][c]);
        __builtin_amdgcn_s_wait_asynccnt(0);
    }
#else
    for (int idx = tid; idx < 16 * D; idx += 256) {
        int r = idx >> 6, c = idx & 63;
        sUi[r][c] = emb[(size_t)sRi[r] * D + c];
        sUj[r][c] = emb[(size_t)sRj[r] * D + c];
    }
#endif
    __syncthreads();

    const int wave = tid >> 5;
    const int lane = tid & 31;
    const int hi   = lane >> 4;     // half-wave: selects K-pair (A/B) and row group (C/D)
    const int ln   = lane & 15;     // A: row M ; B/C/D: column N

    // A fragments: lane M = ln, component v -> K = 4*kk + 2*hi + v.
    v2f aI[16], aJ[16];
#pragma unroll
    for (int kk = 0; kk < 16; ++kk) {
        int k0 = kk * 4 + hi * 2;
        aI[kk].x = sUi[ln][k0];  aI[kk].y = sUi[ln][k0 + 1];
        aJ[kk].x = sUj[ln][k0];  aJ[kk].y = sUj[ln][k0 + 1];
    }

    // Hoist adj row base pointers + k targets out of the hot loop (kills per-load
    // ds_load + v_mul_u64 address rebuilds seen in round-1 codegen).
    const float* aRowI[8];
    const float* aRowJ[8];
    int kTgt[8];
#pragma unroll
    for (int r = 0; r < 8; ++r) {
        int m = r + hi * 8;
        aRowI[r] = adj + (size_t)sRi[m] * (size_t)N;
        aRowJ[r] = adj + (size_t)sRj[m] * (size_t)N;
        kTgt[r]  = sK[m];
    }

    float lsum[8];
#pragma unroll
    for (int r = 0; r < 8; ++r) lsum[r] = 0.f;

    for (int cb = colStart; cb < colStart + colsPerSplit; cb += CPI) {
        __syncthreads();
        // ---- stage theta tile (CPI x 64) into LDS ----
#if USE_ASYNC
        for (int idx = tid; idx < CPI * (D / 4); idx += 256) {
            int r = idx >> 4;             // 16 chunks per row
            int c = (idx & 15) * 4;
            async_copy16(theta + (size_t)(cb + r) * D + c, &sTh[r][c]);
        }
        __builtin_amdgcn_s_wait_asynccnt(0);
#else
        for (int idx = tid; idx < CPI * D; idx += 256) {
            int r = idx >> 6, c = idx & 63;
            sTh[r][c] = theta[(size_t)(cb + r) * D + c];
        }
#endif
        __syncthreads();

        // GEMM: acc[m][n] = sum_k U[m][k] * theta[col_n][k]
        v8f accI = {}; v8f accJ = {};
        const int bcol = wave * 16 + ln;   // theta row (= output column) in tile
#pragma unroll
        for (int kk = 0; kk < 16; ++kk) {
            int k0 = kk * 4 + hi * 2;
            v2f b;
            b.x = sTh[bcol][k0];
            b.y = sTh[bcol][k0 + 1];
            accI = __builtin_amdgcn_wmma_f32_16x16x4_f32(false, aI[kk], false, b,
                                                         (short)0, accI, false, false);
            accJ = __builtin_amdgcn_wmma_f32_16x16x4_f32(false, aJ[kk], false, b,
                                                         (short)0, accJ, false, false);
        }

        // Epilogue on accumulator layout: lane -> col (ln), VGPR r -> row r + 8*hi
        const int col = cb + wave * 16 + ln;
        const float dg = diag[col];
#pragma unroll
        for (int r = 0; r < 8; ++r) {
            float ai = aRowI[r][col];
            float aj = aRowJ[r][col];
            float dot = ai * (dg - accI[r]) + aj * (dg - accJ[r]);
            if (ai > 0.f && aj > 0.f) {
                // log(1 - sigmoid(dot)) = -softplus(dot); dots<-100 cutoff -> term 0
                lsum[r] -= softplusf(dot);
            }
            if (col == kTgt[r]) dotk[sBase + r + hi * 8] = dot;  // unique writer
        }
    }

    // reduce each row's partial across the 16 lanes of its half-wave
    for (int off = 1; off <= 8; off <<= 1) {
#pragma unroll
        for (int r = 0; r < 8; ++r) lsum[r] += __shfl_xor(lsum[r], off, 32);
    }
    if (ln == 0) {
#pragma unroll
        for (int r = 0; r < 8; ++r)
            atomicAdd(&logsum[sBase + r + hi * 8], lsum[r]);
    }
}

// ---------------- per-sample C / log_p terms ----------------
__global__ void k_finish(const float* __restrict__ logsum, const float* __restrict__ dotk,
                         const int* __restrict__ cond, float* acc_tr, int S) {
    int s = blockIdx.x * 256 + threadIdx.x;
    float v = 0.f;
    if (s < S) {
        int c = cond[s];
        if (c != -1) {
            float dk = dotk[s];
            float logp = softplusf(-dk);
            float vest;
            if (c == 1) {
                float C1 = 1.f - expf(logsum[s]);                 // 1 - prod(1-P)
                float C0 = (dk < -100.f) ? 0.f : 1.f / (1.f + expf(-dk));
                vest = 1.f - C0 / (C1 + 1e-6f);
            } else {
                vest = dk;
            }
            v = logp + vest;
        }
    }
    __shared__ float red[8];
    for (int off = 16; off >= 1; off >>= 1) v += __shfl_xor(v, off, 32);
    int wave = threadIdx.x >> 5, lane = threadIdx.x & 31;
    if (lane == 0) red[wave] = v;
    __syncthreads();
    if (threadIdx.x == 0) {
        float t = 0.f;
        for (int w = 0; w < 8; ++w) t += red[w];
        atomicAdd(acc_tr, t);
    }
}

// ---------------- combine ----------------
__global__ void k_combine(const float* acc, float* out, float Sf) {
    if (threadIdx.x == 0)
        out[0] = acc[0] + 0.1f * acc[2] + 0.1f * (Sf * acc[1]);
}

extern "C" void kernel_launch(void* const* d_in, const int* in_sizes, int n_in,
                              void* d_out, int out_size, void* d_ws, size_t ws_size,
                              hipStream_t stream) {
    const float* emb_t = (const float*)d_in[0];
    const float* emb_p = (const float*)d_in[1];
    const float* theta = (const float*)d_in[2];
    const float* adj   = (const float*)d_in[3];
    const int* jI   = (const int*)d_in[4];
    const int* kI   = (const int*)d_in[5];
    const int* nI   = (const int*)d_in[6];
    const int* iI   = (const int*)d_in[7];
    const int* cond = (const int*)d_in[8];

    const int S = in_sizes[4];
    const int N = in_sizes[0] / D;

    float* ws     = (float*)d_ws;
    float* acc    = ws;          // [0]=L_sh  [1]=smooth-sum  [2]=L_tr
    float* diag   = ws + 3;      // N
    float* logsum = diag + N;    // S
    float* dotk   = logsum + S;  // S (always fully overwritten)

    int zN = 3 + N + 2 * S;
    k_init   <<<(zN + 255) / 256, 256, 0, stream>>>(ws, zN);
    k_prep   <<<(N + 7) / 8,      256, 0, stream>>>(emb_t, emb_p, theta, diag, acc + 1, N);
    k_triplet<<<(S + 7) / 8,      256, 0, stream>>>(emb_t, jI, kI, nI, acc, S);
    k_main   <<<(S / 16) * SPLIT, 256, 0, stream>>>(emb_t, theta, adj, diag,
                                                    iI, jI, kI, logsum, dotk, N);
    k_finish <<<(S + 255) / 256,  256, 0, stream>>>(logsum, dotk, cond, acc + 2, S);
    k_combine<<<1, 32, 0, stream>>>(acc, (float*)d_out, (float)S);
}